// SimpleGCNClassifier_35107062678357
// MI455X (gfx1250) — compile-verified
//
#include <hip/hip_runtime.h>

// ---------------------------------------------------------------------------
// SimpleGCNClassifier for MI455X (gfx1250, wave32)
//
// Strategy: transform-before-aggregate (valid since aggregation is linear and
// inv_deg is a per-row scalar), fp32 WMMA 16x16x4 for both GEMMs (exact fp32
// semantics; GEMM cost is negligible vs the 3.3 GB of edge traffic), and
// L2-resident float atomics for the segment sums (the 25.6 MB accumulation
// buffers fit easily in the 192 MB L2).
// ---------------------------------------------------------------------------

typedef float v2f __attribute__((ext_vector_type(2)));
typedef float v8f __attribute__((ext_vector_type(8)));

#define D_IN_   128
#define D_HID_  64

// ---------------------------------------------------------------- zero fill
__global__ void gcn_zero_kernel(float* __restrict__ p, size_t n) {
  size_t i = (size_t)blockIdx.x * blockDim.x + threadIdx.x;
  if (i < n) p[i] = 0.0f;
}

// ---------------------------------------------------------------- in-degree
__global__ void gcn_deg_kernel(const int* __restrict__ dst,
                               float* __restrict__ deg, int nEdges) {
  int e = blockIdx.x * blockDim.x + threadIdx.x;
  if (e < nEdges) atomicAdd(&deg[dst[e]], 1.0f);
}

// -------------------------------------------------- C[M x 64] = A[M x K] @ B[K x 64]
// fp32 WMMA 16x16x4. Requires M % 16 == 0, K % 4 == 0. One wave per 16x16 C tile.
// A-layout (ISA 7.12.2): lane<16 -> K=kb+{0,1}; lane>=16 -> K=kb+{2,3}; M = lane&15.
// B mirrored; C/D: VGPR r -> M = r + (lane>=16 ? 8 : 0), N = lane&15.
__global__ void gcn_gemm_wmma_kernel(const float* __restrict__ A,
                                     const float* __restrict__ B,
                                     float* __restrict__ C, int M, int K) {
  const int N = D_HID_;
  int wave   = (int)((blockIdx.x * blockDim.x + threadIdx.x) >> 5);
  int nTiles = N / 16;                       // 4
  int mTile  = wave / nTiles;
  int nTile  = wave - mTile * nTiles;
  if (mTile * 16 >= M) return;               // wave-uniform; grid sized exactly anyway

  int lane = threadIdx.x & 31;
  int half = lane >> 4;                      // 0: K pair {0,1}; 1: K pair {2,3}
  int l16  = lane & 15;

  const float* arow = A + (size_t)(mTile * 16 + l16) * K;
  const float* bcol = B + nTile * 16 + l16;

  v8f c = {0.f, 0.f, 0.f, 0.f, 0.f, 0.f, 0.f, 0.f};
  for (int kb = 0; kb < K; kb += 4) {
    int ka = kb + half * 2;
    v2f a, b;
    a.x = arow[ka];
    a.y = arow[ka + 1];
    b.x = bcol[(size_t)ka * N];
    b.y = bcol[(size_t)(ka + 1) * N];
    // 8 args: (neg_a, A, neg_b, B, c_mod, C, reuse_a, reuse_b)
    c = __builtin_amdgcn_wmma_f32_16x16x4_f32(false, a, false, b,
                                              (short)0, c, false, false);
  }

  float* crow = C + (size_t)(mTile * 16 + half * 8) * N + nTile * 16 + l16;
#pragma unroll
  for (int r = 0; r < 8; ++r) crow[(size_t)r * N] = c[r];
}

// ------------------------------------------- edge scatter-add: S[dst] += Y[src]
// 16 threads per edge; each thread: one float4 gather + 4 f32 atomics (L2-resident).
__global__ void gcn_agg_kernel(const int* __restrict__ src,
                               const int* __restrict__ dst,
                               const float* __restrict__ Y,
                               float* __restrict__ S, int nEdges) {
  unsigned int t = blockIdx.x * blockDim.x + threadIdx.x;
  int e   = (int)(t >> 4);
  int sub = (int)(t & 15u);
  if (e >= nEdges) return;
  int s = src[e];
  int d = dst[e];
  const float4 v = ((const float4*)(Y + (size_t)s * D_HID_))[sub];
  float* srow = S + (size_t)d * D_HID_ + sub * 4;
  atomicAdd(srow + 0, v.x);
  atomicAdd(srow + 1, v.y);
  atomicAdd(srow + 2, v.z);
  atomicAdd(srow + 3, v.w);
}

// ---------------------------------- H = relu(S * inv_deg + bias), elementwise
__global__ void gcn_finalize_kernel(const float* __restrict__ S,
                                    const float* __restrict__ deg,
                                    const float* __restrict__ bias,
                                    float* __restrict__ H, int nNodes) {
  int t = blockIdx.x * blockDim.x + threadIdx.x;
  if (t >= nNodes * D_HID_) return;
  int n = t >> 6;           // / 64
  int d = t & 63;
  float inv = 1.0f / fmaxf(deg[n], 1.0f);
  float v   = S[t] * inv + bias[d];
  H[t] = fmaxf(v, 0.0f);
}

// ---------------------------------- partial column sums of H -> atomicAdd acc[64]
__global__ void gcn_reduce_kernel(const float* __restrict__ H,
                                  float* __restrict__ acc, int nNodes) {
  int d         = threadIdx.x & 63;
  int rowsPerBk = blockDim.x >> 6;                     // 4 with 256 threads
  int row       = blockIdx.x * rowsPerBk + (threadIdx.x >> 6);
  int stride    = gridDim.x * rowsPerBk;
  float s = 0.0f;
  for (int n = row; n < nNodes; n += stride) s += H[(size_t)n * D_HID_ + d];
  atomicAdd(&acc[d], s);
}

// ---------------------------------- out[0] = (acc/nNodes) . Wc + bc
__global__ void gcn_final_kernel(const float* __restrict__ acc,
                                 const float* __restrict__ Wc,
                                 const float* __restrict__ bc,
                                 float* __restrict__ out, int nNodes) {
  if (threadIdx.x == 0 && blockIdx.x == 0) {
    float invN = 1.0f / (float)nNodes;
    float s = 0.0f;
    for (int d = 0; d < D_HID_; ++d) s += acc[d] * invN * Wc[d];
    out[0] = s + bc[0];
  }
}

// ---------------------------------------------------------------------------
extern "C" void kernel_launch(void* const* d_in, const int* in_sizes, int n_in,
                              void* d_out, int out_size, void* d_ws, size_t ws_size,
                              hipStream_t stream) {
  const float* feat = (const float*)d_in[0];
  const int*   src  = (const int*)d_in[1];
  const int*   dst  = (const int*)d_in[2];
  const float* W1   = (const float*)d_in[3];
  const float* b1   = (const float*)d_in[4];
  const float* W2   = (const float*)d_in[5];
  const float* b2   = (const float*)d_in[6];
  const float* Wc   = (const float*)d_in[7];
  const float* bc   = (const float*)d_in[8];
  float*       out  = (float*)d_out;

  const int nNodes = in_sizes[0] / D_IN_;   // 100000 (multiple of 16)
  const int nEdges = in_sizes[1];           // 3200000

  // Workspace layout (floats): [deg nNodes][acc 64][buf0][buf1][buf2], each buf nNodes*64
  float*       w    = (float*)d_ws;
  const size_t NM   = (size_t)nNodes * D_HID_;
  float* deg  = w;
  float* acc  = w + nNodes;
  float* buf0 = acc + 64;           // Y1 -> h1 -> h2
  float* buf1 = buf0 + NM;          // S1 -> Y2
  float* buf2 = buf1 + NM;          // S2

  const int TB = 256;
  const size_t totalZero = (size_t)nNodes + 64 + 3 * NM;

  // 1) zero deg/acc and all accumulation buffers
  gcn_zero_kernel<<<(unsigned)((totalZero + TB - 1) / TB), TB, 0, stream>>>(w, totalZero);

  // 2) in-degree
  gcn_deg_kernel<<<(nEdges + TB - 1) / TB, TB, 0, stream>>>(dst, deg, nEdges);

  // 3) Y1 = feat @ W1   [nNodes,128]x[128,64]  (WMMA f32 16x16x4)
  const int wavesPerGemm = (nNodes / 16) * (D_HID_ / 16);   // 25000
  const int gemmBlocks   = (wavesPerGemm * 32 + TB - 1) / TB;
  gcn_gemm_wmma_kernel<<<gemmBlocks, TB, 0, stream>>>(feat, W1, buf0, nNodes, D_IN_);

  // 4) S1 = segment_sum(Y1[src] -> dst)
  const long long aggThreads = (long long)nEdges * 16;
  const int aggBlocks = (int)((aggThreads + TB - 1) / TB);
  gcn_agg_kernel<<<aggBlocks, TB, 0, stream>>>(src, dst, buf0, buf1, nEdges);

  // 5) h1 = relu(S1 * inv_deg + b1) -> buf0
  const int elemBlocks = (nNodes * D_HID_ + TB - 1) / TB;
  gcn_finalize_kernel<<<elemBlocks, TB, 0, stream>>>(buf1, deg, b1, buf0, nNodes);

  // 6) Y2 = h1 @ W2   [nNodes,64]x[64,64] -> buf1
  gcn_gemm_wmma_kernel<<<gemmBlocks, TB, 0, stream>>>(buf0, W2, buf1, nNodes, D_HID_);

  // 7) S2 = segment_sum(Y2[src] -> dst) -> buf2
  gcn_agg_kernel<<<aggBlocks, TB, 0, stream>>>(src, dst, buf1, buf2, nEdges);

  // 8) h2 = relu(S2 * inv_deg + b2) -> buf0
  gcn_finalize_kernel<<<elemBlocks, TB, 0, stream>>>(buf2, deg, b2, buf0, nNodes);

  // 9) column sums of h2
  gcn_reduce_kernel<<<512, TB, 0, stream>>>(buf0, acc, nNodes);

  // 10) out = (acc / nNodes) @ Wc + bc
  gcn_final_kernel<<<1, 64, 0, stream>>>(acc, Wc, bc, out, nNodes);
}